// DifferentiableRenderer_2173253452332
// MI455X (gfx1250) — compile-verified
//
#include <hip/hip_runtime.h>

// Problem constants (match reference)
#define H_IMG   128
#define W_IMG   128
#define HW      16384      // H*W pixels
#define NPTS    16384
#define KSEL    16
#define FEATD   64

#define CHUNK     32       // points scored per barrier round (2 WMMA sub-tiles)
#define ROWSTRIDE 33       // LDS row stride in floats (odd -> conflict-free scans)

typedef float v2f __attribute__((ext_vector_type(2)));
typedef float v8f __attribute__((ext_vector_type(8)));

// ---------------------------------------------------------------------------
// Kernel 1: project points, build WMMA B-operand rows:
//   ptsB[n] = { -2u, -2v, u^2+v^2, depth }
// Score GEMM:  [x y 1 0] . [-2u -2v u^2+v^2 depth]^T = -2xu -2yv + u^2+v^2
// (depth rides in the K=3 slot; A's K=3 entry is 0 so it contributes nothing)
// ---------------------------------------------------------------------------
__global__ void project_kernel(const float* __restrict__ pts,
                               const float* __restrict__ intr,
                               float4* __restrict__ ptsB) {
    int n = blockIdx.x * blockDim.x + threadIdx.x;
    if (n >= NPTS) return;
    float fx = intr[0], cx = intr[2], fy = intr[4], cy = intr[5];
    float x = pts[n * 3 + 0], y = pts[n * 3 + 1], z = pts[n * 3 + 2];
    bool valid = z > 1e-6f;
    float sz = valid ? z : 1.0f;
    float u = x * fx / sz + cx;
    float v = y * fy / sz + cy;
    if (!valid) { u = 1e9f; v = 1e9f; }   // FAR -> huge score, never selected
    float4 b;
    b.x = -2.0f * u;
    b.y = -2.0f * v;
    b.z = u * u + v * v;                  // <= 2e18, fine in f32
    b.w = z;
    ptsB[n] = b;
}

// ---------------------------------------------------------------------------
// Kernel 2: WMMA-scored KNN + splat. 256 threads = 8 waves; wave w scores
// pixel tiles 2w,2w+1 (16 pixels each) against 32-point chunks with
// v_wmma_f32_16x16x4_f32, stages scores in LDS, every thread keeps a
// register-resident sorted top-16 for its pixel, then splats.
// ---------------------------------------------------------------------------
__global__ __launch_bounds__(256, 2)
void knn_render_kernel(const float4* __restrict__ ptsB,
                       const float* __restrict__ colors,
                       const float* __restrict__ feats,
                       float* __restrict__ out_depth,
                       float* __restrict__ out_color,
                       float* __restrict__ out_feat,
                       float* __restrict__ out_mask) {
    __shared__ float sc[256 * ROWSTRIDE];   // 33792 B

    const int t       = threadIdx.x;
    const int lane    = t & 31;
    const int laneLow = lane & 15;
    const bool hiHalf = lane >= 16;
    const int wv      = t >> 5;
    const int wgbase  = blockIdx.x * 256;

    // --- A fragments (16x4 f32: lanes 0-15 hold K=0,1; lanes 16-31 K=2,3) ---
    v2f a0, a1;
    {
        int p0 = wgbase + (2 * wv)     * 16 + laneLow;
        int p1 = wgbase + (2 * wv + 1) * 16 + laneLow;
        float x0 = (float)(p0 % W_IMG) + 0.5f, y0 = (float)(p0 / W_IMG) + 0.5f;
        float x1 = (float)(p1 % W_IMG) + 0.5f, y1 = (float)(p1 / W_IMG) + 0.5f;
        a0[0] = hiHalf ? 1.0f : x0;  a0[1] = hiHalf ? 0.0f : y0;
        a1[0] = hiHalf ? 1.0f : x1;  a1[1] = hiHalf ? 0.0f : y1;
    }

    // --- register-resident sorted top-K (ascending score) ---
    float bs[KSEL]; int bi[KSEL];
#pragma unroll
    for (int k = 0; k < KSEL; ++k) { bs[k] = 3.4e38f; bi[k] = 0; }

    const float* bflat = (const float*)ptsB;

    for (int c0 = 0; c0 < NPTS; c0 += CHUNK) {
        __syncthreads();   // LDS reads of previous chunk complete

#pragma unroll
        for (int s = 0; s < CHUNK / 16; ++s) {
            int n0 = c0 + 16 * s;
            // B fragment (4x16 f32): lanes 0-15 -> rows K=0,1 = (-2u,-2v);
            //                        lanes16-31 -> rows K=2,3 = (u^2+v^2, depth)
            const float* src = bflat + (size_t)(n0 + laneLow) * 4 + (hiHalf ? 2 : 0);
            v2f b; b[0] = src[0]; b[1] = src[1];

            v8f c = {};
            c = __builtin_amdgcn_wmma_f32_16x16x4_f32(
                    false, a0, false, b, (short)0, c, false, false);
            int colb = 16 * s + laneLow;
            int rowb = (2 * wv) * 16 + 8 * (lane >> 4);
#pragma unroll
            for (int v = 0; v < 8; ++v)
                sc[(rowb + v) * ROWSTRIDE + colb] = c[v];

            v8f c1 = {};
            c1 = __builtin_amdgcn_wmma_f32_16x16x4_f32(
                    false, a1, false, b, (short)0, c1, false, false);
            int rowb1 = (2 * wv + 1) * 16 + 8 * (lane >> 4);
#pragma unroll
            for (int v = 0; v < 8; ++v)
                sc[(rowb1 + v) * ROWSTRIDE + colb] = c1[v];
        }

        if (c0 + CHUNK < NPTS)   // warm L2/WGP$ for the next chunk
            __builtin_prefetch(bflat + (size_t)(c0 + CHUNK + laneLow) * 4, 0, 0);

        __syncthreads();   // scores visible

        // --- scan: thread t owns LDS row t (pixel wgbase + t) ---
        const float* row = &sc[t * ROWSTRIDE];
#pragma unroll 4
        for (int q = 0; q < CHUNK; ++q) {
            float sv = row[q];
            if (sv < bs[KSEL - 1]) {          // threshold guard (rare path)
                float cs = sv; int ci = c0 + q;
#pragma unroll
                for (int k = 0; k < KSEL; ++k) {
                    if (cs < bs[k]) {
                        float tf = bs[k]; bs[k] = cs; cs = tf;
                        int   ti = bi[k]; bi[k] = ci; ci = ti;
                    }
                }
            }
        }
    }

    // --- finalize: exact distances, Gaussian weights, splat ---
    // NOTE: outputs are permutation-invariant sums over the K selected points,
    // so the reference's depth argsort is a provable no-op -> skipped.
    const int p = wgbase + t;
    const float px = (float)(p % W_IMG) + 0.5f;
    const float py = (float)(p / W_IMG) + 0.5f;

    float wsum = 0.0f, dsum = 0.0f;
    float ca = 0.0f, cb = 0.0f, cc = 0.0f;
    bool anym = false;
    float facc[FEATD];
#pragma unroll
    for (int c = 0; c < FEATD; ++c) facc[c] = 0.0f;

#pragma unroll
    for (int k = 0; k < KSEL; ++k) {
        int i = bi[k];
        float4 b = ptsB[i];
        float u = -0.5f * b.x, v = -0.5f * b.y, z = b.w;
        float dx = px - u, dy = py - v;
        float d2 = dx * dx + dy * dy;        // exact (no GEMM cancellation)
        bool m = d2 < 4.0f;                  // dist < RADIUS
        anym |= m;
        float w = m ? __expf(-d2) : 0.0f;    // exp(-dist^2/sigma^2), sigma=1
        wsum += w;
        dsum += z * w;
        ca += colors[i * 3 + 0] * w;
        cb += colors[i * 3 + 1] * w;
        cc += colors[i * 3 + 2] * w;
        const float4* f = (const float4*)(feats + (size_t)i * FEATD);
#pragma unroll
        for (int c = 0; c < FEATD / 4; ++c) {
            float4 fv = f[c];
            facc[4 * c + 0] += fv.x * w;
            facc[4 * c + 1] += fv.y * w;
            facc[4 * c + 2] += fv.z * w;
            facc[4 * c + 3] += fv.w * w;
        }
    }

    float inv = 1.0f / (wsum + 1e-10f);
    out_depth[p] = dsum * inv;
    out_color[p * 3 + 0] = ca * inv;
    out_color[p * 3 + 1] = cb * inv;
    out_color[p * 3 + 2] = cc * inv;
    float4* of = (float4*)(out_feat + (size_t)p * FEATD);
#pragma unroll
    for (int c = 0; c < FEATD / 4; ++c) {
        float4 o;
        o.x = facc[4 * c + 0] * inv;
        o.y = facc[4 * c + 1] * inv;
        o.z = facc[4 * c + 2] * inv;
        o.w = facc[4 * c + 3] * inv;
        of[c] = o;
    }
    out_mask[p] = anym ? 1.0f : 0.0f;
}

// ---------------------------------------------------------------------------
extern "C" void kernel_launch(void* const* d_in, const int* in_sizes, int n_in,
                              void* d_out, int out_size, void* d_ws, size_t ws_size,
                              hipStream_t stream) {
    const float* pts  = (const float*)d_in[0];   // (N,3)
    const float* cols = (const float*)d_in[1];   // (N,3)
    const float* fts  = (const float*)d_in[2];   // (N,64)
    const float* intr = (const float*)d_in[3];   // (3,3)

    float* out       = (float*)d_out;
    float* out_depth = out;                                   // 16384
    float* out_color = out + HW;                              // 49152
    float* out_feat  = out + HW + HW * 3;                     // 1048576
    float* out_mask  = out + HW + HW * 3 + HW * FEATD;        // 16384

    float4* ptsB = (float4*)d_ws;                             // 256 KB scratch

    project_kernel<<<NPTS / 256, 256, 0, stream>>>(pts, intr, ptsB);
    knn_render_kernel<<<HW / 256, 256, 0, stream>>>(
        ptsB, cols, fts, out_depth, out_color, out_feat, out_mask);
}